// MatrixModel_12884901888368
// MI455X (gfx1250) — compile-verified
//
#include <hip/hip_runtime.h>

typedef __attribute__((ext_vector_type(2))) float v2f;
typedef __attribute__((ext_vector_type(8))) float v8f;

#define NMAT 512
#define NWG 16
#define ROWS_PER_WG 32
#define TPB 256
#define APITCH 516   // row pitch in floats: stride 516*4B -> banks m*4 mod 64, conflict-free 16-row gathers
#define NITER 1024   // 2 * num_terms

__global__ void sinkhorn_init(unsigned* counter) { *counter = 0u; }

__global__ __launch_bounds__(TPB)
void sinkhorn_kernel(const float* __restrict__ weight, float* __restrict__ out,
                     float* __restrict__ p, unsigned* __restrict__ counter)
{
    extern __shared__ float lds[];
    float* A    = lds;                          // [32][516]
    float* cvec = lds + ROWS_PER_WG * APITCH;   // [512]
    float* rvec = cvec + NMAT;                  // [32]
    float* yscr = rvec + ROWS_PER_WG;           // [2 row-blocks][4 K-quarters][16]

    const int tid  = threadIdx.x;
    const int g    = blockIdx.x;
    const int row0 = g * ROWS_PER_WG;

    // Stage |weight| slice into LDS; init c = ones.
    for (int idx = tid; idx < ROWS_PER_WG * NMAT; idx += TPB) {
        int i = idx >> 9, j = idx & (NMAT - 1);
        A[i * APITCH + j] = fabsf(weight[(size_t)(row0 + i) * NMAT + j]);
    }
    for (int j = tid; j < NMAT; j += TPB) cvec[j] = 1.0f;
    __syncthreads();

    const int wave = tid >> 5, lane = tid & 31;
    const int mb   = (wave & 1) << 4;        // row block within slice: 0 or 16
    const int q    = wave >> 1;              // K quarter: 0..3 (128 K each)
    const int mrow = mb + (lane & 15);       // my A-matrix row (ISA: A rows in lanes, K split by lane half)
    const int koff = (lane >> 4) << 1;       // lanes 0-15: K {0,1}; lanes 16-31: K {2,3}

    for (int t = 0; t < NITER; ++t) {
        // ---- Phase 1: y = A*c via fp32 WMMA 16x16x4 (B = c broadcast across N) ----
        v8f acc = {};
        const float* Arow = A + mrow * APITCH;
        for (int k0 = q * 128; k0 < q * 128 + 128; k0 += 4) {
            int ka = k0 + koff;
            v2f a; a.x = Arow[ka];  a.y = Arow[ka + 1];
            v2f b; b.x = cvec[ka];  b.y = cvec[ka + 1];
            acc = __builtin_amdgcn_wmma_f32_16x16x4_f32(
                      false, a, false, b, (short)0, acc, false, false);
        }
        // D layout: VGPR v, lane 0 -> (M=v, N=0); lane 16 -> (M=v+8, N=0). Column 0 = row sums.
        if ((lane & 15) == 0) {
            float* ys = yscr + (((mb >> 4) << 2) + q) * 16 + ((lane >> 4) << 3);
            #pragma unroll
            for (int v = 0; v < 8; ++v) ys[v] = acc[v];
        }
        __syncthreads();
        if (tid < ROWS_PER_WG) {
            float* ys = yscr + ((tid >> 4) << 2) * 16;
            int m = tid & 15;
            float y = ys[m] + ys[16 + m] + ys[32 + m] + ys[48 + m];
            rvec[tid] = 1.0f / y;
        }
        __syncthreads();

        // ---- Phase 2: partial z_j = sum_i A[i][j] * r[i] -> global partial buffer ----
        float* pbuf = p + (size_t)(t & 1) * NWG * NMAT + (size_t)g * NMAT;
        for (int j = tid; j < NMAT; j += TPB) {
            float z = 0.f;
            #pragma unroll
            for (int i = 0; i < ROWS_PER_WG; ++i)
                z = fmaf(A[i * APITCH + j], rvec[i], z);
            pbuf[j] = z;
        }

        // ---- Grid barrier (monotonic counter; one sync/iter thanks to double buffering) ----
        __threadfence();                 // release: my partial stores reach device scope
        __syncthreads();
        if (tid == 0) {
            __hip_atomic_fetch_add(counter, 1u, __ATOMIC_RELEASE, __HIP_MEMORY_SCOPE_AGENT);
            unsigned target = (unsigned)NWG * (unsigned)(t + 1);
            while (__hip_atomic_load(counter, __ATOMIC_ACQUIRE, __HIP_MEMORY_SCOPE_AGENT) < target)
                __builtin_amdgcn_s_sleep(1);
        }
        __syncthreads();
        __threadfence();                 // acquire: refresh WGP read cache

        // ---- Phase 3: reduce 16 partials, c = 1/z (computed redundantly per WG) ----
        const float* pall = p + (size_t)(t & 1) * NWG * NMAT;
        for (int j = tid; j < NMAT; j += TPB) {
            float z = 0.f;
            #pragma unroll
            for (int g2 = 0; g2 < NWG; ++g2) z += pall[g2 * NMAT + j];
            cvec[j] = 1.0f / z;
        }
        __syncthreads();
    }

    // ---- Output: W = diag(r) * A * diag(c) ----
    for (int idx = tid; idx < ROWS_PER_WG * NMAT; idx += TPB) {
        int i = idx >> 9, j = idx & (NMAT - 1);
        out[(size_t)(row0 + i) * NMAT + j] = rvec[i] * A[i * APITCH + j] * cvec[j];
    }
}

extern "C" void kernel_launch(void* const* d_in, const int* in_sizes, int n_in,
                              void* d_out, int out_size, void* d_ws, size_t ws_size,
                              hipStream_t stream) {
    const float* weight = (const float*)d_in[0];
    float* out = (float*)d_out;
    float* p = (float*)d_ws;                                   // 2 * 16 * 512 floats = 64 KB
    unsigned* counter = (unsigned*)((char*)d_ws + (size_t)2 * NWG * NMAT * sizeof(float));

    sinkhorn_init<<<1, 1, 0, stream>>>(counter);

    size_t shmem = (size_t)(ROWS_PER_WG * APITCH + NMAT + ROWS_PER_WG + 2 * 4 * 16) * sizeof(float);
    sinkhorn_kernel<<<NWG, TPB, shmem, stream>>>(weight, out, p, counter);
}